// BANLayer_51745765983005
// MI455X (gfx1250) — compile-verified
//
#include <hip/hip_runtime.h>
#include <hip/hip_bf16.h>

typedef __attribute__((ext_vector_type(2))) float v2f;
typedef __attribute__((ext_vector_type(8))) float v8f;

#define B_    32
#define NV_   256
#define NQ_   512
#define V_DIM 128
#define Q_DIM 768
#define H_DIM 512
#define H_OUT 8
#define KK    3
#define KD    1536   // H_DIM * K

// ---------------------------------------------------------------------------
// Generic 16x16-tile fp32 GEMM with bias + ReLU:  Y[rows,ncols] = relu(X*W^T+b)
// X: [rows,kdim] row-major, W: [ncols,kdim] row-major.
// One wave per 16x16 output tile; K stepped by 4 (V_WMMA_F32_16X16X4_F32).
// A-frag (16x4): lane l holds A[l%16, k + 2*(l/16) + {0,1}] -> packed v2f load.
// B-frag (4x16): lane l holds B[k + 2*(l/16) + {0,1}, l%16] = W[l%16, ...].
// ---------------------------------------------------------------------------
__global__ __launch_bounds__(256) void proj_gemm_relu(
    const float* __restrict__ X, const float* __restrict__ W,
    const float* __restrict__ bias, float* __restrict__ Y,
    int rows, int kdim, int ncols)
{
    int wave = blockIdx.x * (blockDim.x >> 5) + (threadIdx.x >> 5);
    int lane = threadIdx.x & 31;
    int tilesN = ncols >> 4;
    int tm = wave / tilesN;
    int tn = wave - tm * tilesN;
    if (tm * 16 >= rows) return;                 // uniform per wave

    int lmod = lane & 15;
    int lhi  = lane >> 4;
    int koff = lhi << 1;

    const float* xrow = X + (size_t)(tm * 16 + lmod) * kdim;
    const float* wrow = W + (size_t)(tn * 16 + lmod) * kdim;

    v8f acc = {};
    for (int k = 0; k < kdim; k += 4) {
        v2f a = *(const v2f*)(xrow + k + koff);
        v2f b = *(const v2f*)(wrow + k + koff);
        acc = __builtin_amdgcn_wmma_f32_16x16x4_f32(false, a, false, b,
                                                    (short)0, acc, false, false);
    }
    int col = tn * 16 + lmod;
    float bi = bias[col];
#pragma unroll
    for (int r = 0; r < 8; ++r) {
        int row = tm * 16 + r + (lhi << 3);
        float val = acc[r] + bi;
        Y[(size_t)row * ncols + col] = val > 0.f ? val : 0.f;
    }
}

// ---------------------------------------------------------------------------
// att[b,h,v,q] = sum_k (h_mat[h,k]*v_[b,v,k]) * q_[b,q,k] + h_bias[h]
// Block = (v-tile, h, b); 8 waves; each wave: 16v x 64q via 4 accumulators.
// Raw (masked) logits written straight into the att region of d_out.
// ---------------------------------------------------------------------------
__global__ __launch_bounds__(256) void att_logits(
    const float* __restrict__ v_, const float* __restrict__ q_,
    const float* __restrict__ h_mat, const float* __restrict__ h_bias,
    const int* __restrict__ v_mask, const int* __restrict__ q_mask,
    const int* __restrict__ softmax_flag, float* __restrict__ att)
{
    int vt = blockIdx.x;          // NV/16
    int h  = blockIdx.y;          // H_OUT
    int b  = blockIdx.z;          // B
    int wave = threadIdx.x >> 5;
    int lane = threadIdx.x & 31;
    int lmod = lane & 15, lhi = lane >> 4;
    int koff = lhi << 1;
    int v0 = vt * 16;
    int q0 = wave * 64;

    const float* arow = v_ + (size_t)(b * NV_ + v0 + lmod) * KD;
    const float* hrow = h_mat + (size_t)h * KD;
    const float* qb   = q_ + (size_t)b * NQ_ * KD;
    const float* qp[4] = {
        qb + (size_t)(q0 +  0 + lmod) * KD,
        qb + (size_t)(q0 + 16 + lmod) * KD,
        qb + (size_t)(q0 + 32 + lmod) * KD,
        qb + (size_t)(q0 + 48 + lmod) * KD };

    v8f acc[4] = {{}, {}, {}, {}};
    for (int k = 0; k < KD; k += 4) {
        v2f a = *(const v2f*)(arow + k + koff) * *(const v2f*)(hrow + k + koff);
#pragma unroll
        for (int j = 0; j < 4; ++j) {
            v2f bj = *(const v2f*)(qp[j] + k + koff);
            acc[j] = __builtin_amdgcn_wmma_f32_16x16x4_f32(false, a, false, bj,
                                                           (short)0, acc[j], false, false);
        }
    }

    float hb = h_bias[h];
    int sm = *softmax_flag;
    float* outp = att + (size_t)(b * H_OUT + h) * NV_ * NQ_;
#pragma unroll
    for (int j = 0; j < 4; ++j) {
        int qc = q0 + j * 16 + lmod;
        int qm = q_mask[b * NQ_ + qc];
#pragma unroll
        for (int r = 0; r < 8; ++r) {
            int vr = v0 + r + (lhi << 3);
            float val = acc[j][r] + hb;
            if (sm && (v_mask[b * NV_ + vr] == 0 || qm == 0)) val = -1e9f;
            outp[(size_t)vr * NQ_ + qc] = val;
        }
    }
}

// ---------------------------------------------------------------------------
// Per-(b,h) softmax statistics over the flattened 131072-element (v,q) grid.
// ---------------------------------------------------------------------------
#define STAT_T 1024
__global__ __launch_bounds__(STAT_T) void softmax_stats(
    const float* __restrict__ att, const int* __restrict__ softmax_flag,
    float* __restrict__ stats)
{
    if (*softmax_flag == 0) return;
    __shared__ float smem[STAT_T];
    int bh = blockIdx.x;
    const float* p = att + (size_t)bh * (NV_ * NQ_);
    int t = threadIdx.x;

    float m = -3.4e38f;
    for (int i = t; i < NV_ * NQ_; i += STAT_T) m = fmaxf(m, p[i]);
    smem[t] = m; __syncthreads();
    for (int s = STAT_T / 2; s > 0; s >>= 1) {
        if (t < s) smem[t] = fmaxf(smem[t], smem[t + s]);
        __syncthreads();
    }
    m = smem[0]; __syncthreads();

    float acc = 0.f;
    for (int i = t; i < NV_ * NQ_; i += STAT_T) acc += __expf(p[i] - m);
    smem[t] = acc; __syncthreads();
    for (int s = STAT_T / 2; s > 0; s >>= 1) {
        if (t < s) smem[t] += smem[t + s];
        __syncthreads();
    }
    if (t == 0) { stats[bh * 2] = m; stats[bh * 2 + 1] = smem[0]; }
}

// ---------------------------------------------------------------------------
// Normalize att in place (exp(x-m)/s + 1e-10) and produce att_sum = sum_h att.
// One thread per (b,v,q); loops the 8 heads so att_sum needs no atomics.
// ---------------------------------------------------------------------------
__global__ __launch_bounds__(256) void softmax_norm(
    float* __restrict__ att, const int* __restrict__ softmax_flag,
    const float* __restrict__ stats, float* __restrict__ att_sum)
{
    size_t t = (size_t)blockIdx.x * blockDim.x + threadIdx.x;
    const size_t plane = (size_t)NV_ * NQ_;
    if (t >= (size_t)B_ * plane) return;
    int b = (int)(t / plane);
    size_t r = t - (size_t)b * plane;
    int sm = *softmax_flag;

    float total = 0.f;
#pragma unroll
    for (int h = 0; h < H_OUT; ++h) {
        size_t idx = ((size_t)(b * H_OUT + h)) * plane + r;
        float x = att[idx];
        if (sm) {
            float m = stats[(b * H_OUT + h) * 2];
            float s = stats[(b * H_OUT + h) * 2 + 1];
            x = __expf(x - m) / s + 1e-10f;
            att[idx] = x;
        }
        total += x;
    }
    att_sum[t] = total;
}

// ---------------------------------------------------------------------------
// fusion[b,k] = sum_q q_[b,q,k] * (sum_v att_sum[b,v,q] * v_[b,v,k])
// WMMA tile T[16q,16k] reduced over v (steps of 4); epilogue dots each column of T
// with q_ and atomically accumulates into fusion.
// ---------------------------------------------------------------------------
__global__ __launch_bounds__(256) void fusion_kernel(
    const float* __restrict__ v_, const float* __restrict__ q_,
    const float* __restrict__ att_sum, float* __restrict__ fusion)
{
    const int tilesK = KD / 16;   // 96
    const int tilesQ = NQ_ / 16;  // 32
    int wave = blockIdx.x * (blockDim.x >> 5) + (threadIdx.x >> 5);
    int lane = threadIdx.x & 31;
    int kt = wave % tilesK;
    int qt = (wave / tilesK) % tilesQ;
    int b  = wave / (tilesK * tilesQ);
    if (b >= B_) return;

    int lmod = lane & 15, lhi = lane >> 4, koff = lhi << 1;
    int q0 = qt * 16, k0 = kt * 16;
    const float* as = att_sum + (size_t)b * NV_ * NQ_;
    const float* vb = v_ + (size_t)b * NV_ * KD;

    v8f acc = {};
    for (int vv = 0; vv < NV_; vv += 4) {
        int vi = vv + koff;
        v2f a  = { as[(size_t)vi * NQ_ + q0 + lmod],
                   as[(size_t)(vi + 1) * NQ_ + q0 + lmod] };
        v2f bb = { vb[(size_t)vi * KD + k0 + lmod],
                   vb[(size_t)(vi + 1) * KD + k0 + lmod] };
        acc = __builtin_amdgcn_wmma_f32_16x16x4_f32(false, a, false, bb,
                                                    (short)0, acc, false, false);
    }

    float s = 0.f;
    const float* qp = q_ + (size_t)b * NQ_ * KD;
#pragma unroll
    for (int r = 0; r < 8; ++r) {
        int qr = q0 + r + (lhi << 3);
        s += acc[r] * qp[(size_t)qr * KD + k0 + lmod];
    }
    atomicAdd(&fusion[b * KD + k0 + lmod], s);
}

// ---------------------------------------------------------------------------
// pooled = group-of-3 sum; BatchNorm with biased batch stats; logits out.
// One thread per channel (512 channels, B=32 batch).
// ---------------------------------------------------------------------------
__global__ __launch_bounds__(256) void bn_pool(
    const float* __restrict__ fusion, const float* __restrict__ gamma,
    const float* __restrict__ beta, float* __restrict__ out)
{
    int c = blockIdx.x * blockDim.x + threadIdx.x;
    if (c >= H_DIM) return;
    float s1 = 0.f, s2 = 0.f;
    for (int b = 0; b < B_; ++b) {
        const float* f = fusion + (size_t)b * KD + 3 * c;
        float p = f[0] + f[1] + f[2];
        s1 += p; s2 += p * p;
    }
    float mean = s1 * (1.f / B_);
    float var  = s2 * (1.f / B_) - mean * mean;
    float inv  = rsqrtf(var + 1e-5f);
    float g = gamma[c], bt = beta[c];
    for (int b = 0; b < B_; ++b) {
        const float* f = fusion + (size_t)b * KD + 3 * c;
        float p = f[0] + f[1] + f[2];
        out[(size_t)b * H_DIM + c] = g * (p - mean) * inv + bt;
    }
}

__global__ void zero_kernel(float* __restrict__ p, int n)
{
    int i = blockIdx.x * blockDim.x + threadIdx.x;
    if (i < n) p[i] = 0.f;
}

// ---------------------------------------------------------------------------
extern "C" void kernel_launch(void* const* d_in, const int* in_sizes, int n_in,
                              void* d_out, int out_size, void* d_ws, size_t ws_size,
                              hipStream_t stream)
{
    const float* v       = (const float*)d_in[0];
    const float* q       = (const float*)d_in[1];
    const int*   v_mask  = (const int*)d_in[2];
    const int*   q_mask  = (const int*)d_in[3];
    const int*   softmax = (const int*)d_in[4];
    const float* v_W     = (const float*)d_in[5];
    const float* v_b     = (const float*)d_in[6];
    const float* q_W     = (const float*)d_in[7];
    const float* q_b     = (const float*)d_in[8];
    const float* h_mat   = (const float*)d_in[9];
    const float* h_bias  = (const float*)d_in[10];
    const float* gamma   = (const float*)d_in[11];
    const float* beta    = (const float*)d_in[12];

    float* out_logits = (float*)d_out;                       // [B, H_DIM]
    float* out_att    = (float*)d_out + (size_t)B_ * H_DIM;  // [B, H_OUT, NV, NQ]

    // workspace layout (floats)
    float* ws = (float*)d_ws;
    float* v_act   = ws;                                        // 12,582,912
    float* q_act   = v_act + (size_t)B_ * NV_ * KD;             // 25,165,824
    float* att_sum = q_act + (size_t)B_ * NQ_ * KD;             //  4,194,304
    float* fusion  = att_sum + (size_t)B_ * NV_ * NQ_;          //     49,152
    float* stats   = fusion + (size_t)B_ * KD;                  //        512

    // 0) zero the fusion accumulator
    zero_kernel<<<(B_ * KD + 255) / 256, 256, 0, stream>>>(fusion, B_ * KD);

    // 1) v_ = relu(v @ v_W^T + v_b)   [8192 x 1536, K=128]
    {
        int rows = B_ * NV_, tiles = (rows / 16) * (KD / 16);
        proj_gemm_relu<<<tiles / 8, 256, 0, stream>>>(v, v_W, v_b, v_act,
                                                      rows, V_DIM, KD);
    }
    // 2) q_ = relu(q @ q_W^T + q_b)   [16384 x 1536, K=768]
    {
        int rows = B_ * NQ_, tiles = (rows / 16) * (KD / 16);
        proj_gemm_relu<<<tiles / 8, 256, 0, stream>>>(q, q_W, q_b, q_act,
                                                      rows, Q_DIM, KD);
    }
    // 3) bilinear attention logits -> att region of d_out (raw, masked)
    {
        dim3 grid(NV_ / 16, H_OUT, B_);
        att_logits<<<grid, 256, 0, stream>>>(v_act, q_act, h_mat, h_bias,
                                             v_mask, q_mask, softmax, out_att);
    }
    // 4) softmax stats per (b,h)
    softmax_stats<<<B_ * H_OUT, STAT_T, 0, stream>>>(out_att, softmax, stats);

    // 5) normalize in place + att_sum = sum over heads
    {
        size_t n = (size_t)B_ * NV_ * NQ_;
        softmax_norm<<<(unsigned)((n + 255) / 256), 256, 0, stream>>>(
            out_att, softmax, stats, att_sum);
    }
    // 6) fusion[b,k]
    {
        int waves = B_ * (NQ_ / 16) * (KD / 16);
        fusion_kernel<<<waves / 8, 256, 0, stream>>>(v_act, q_act, att_sum, fusion);
    }
    // 7) pool-by-3 + batchnorm -> logits
    bn_pool<<<(H_DIM + 255) / 256, 256, 0, stream>>>(fusion, gamma, beta, out_logits);
}